// PDAG_Generator_19224273617537
// MI455X (gfx1250) — compile-verified
//
#include <hip/hip_runtime.h>
#include <hip/hip_bf16.h>

// ---------------- problem constants ----------------
#define BQ      16384     // batch
#define HQ      512       // hidden
#define IDIM    16        // sequential steps
#define K0PAD   20        // layer-0 K padded (17/18 -> 20, zero filled)
#define K0P2    10        // K0PAD/2 pairs
#define H1S     516       // LDS row stride for h1 (pad to dodge bank conflicts)

typedef __attribute__((ext_vector_type(2))) float v2f;
typedef __attribute__((ext_vector_type(8))) float v8f;

__device__ __forceinline__ v8f wmma_f32_4(v2f a, v2f b, v8f c) {
    // V_WMMA_F32_16X16X4_F32 : full fp32 matrix op on CDNA5
    return __builtin_amdgcn_wmma_f32_16x16x4_f32(
        /*neg_a=*/false, a, /*neg_b=*/false, b,
        /*c_mod=*/(short)0, c, /*reuse_a=*/false, /*reuse_b=*/false);
}

// ---------------- prep: theta = mu + softplus(sigma)*noise_d ; kl out ----------------
__global__ void prep_theta_kl(const float* __restrict__ mu,
                              const float* __restrict__ sigma,
                              const float* __restrict__ noise_d,
                              float* __restrict__ theta,
                              float* __restrict__ kl_out) {
    __shared__ float red[4];
    int t = threadIdx.x;               // 0..511 over (4 x 128)
    if (t < 4) red[t] = 0.0f;
    __syncthreads();
    float s  = sigma[t];
    float m  = mu[t];
    float sp = log1pf(expf(s));
    theta[t] = m + sp * noise_d[t];
    float kl = 1.0f + logf(sp * sp) - m * m - sp * sp;
    atomicAdd(&red[t >> 7], kl);
    __syncthreads();
    if (t < 4) kl_out[t] = -red[t];
}

// ---------------- prep: inputs_c = input_c @ cnet_w^T ; inputs_d = input_d @ theta ----
__global__ void prep_inputs(const float* __restrict__ input_c,
                            const float* __restrict__ input_d,
                            const float* __restrict__ cnet_w,
                            const float* __restrict__ theta,
                            float* __restrict__ xc,
                            float* __restrict__ xd) {
    int idx = blockIdx.x * 256 + threadIdx.x;      // over B*128
    int n = idx & 127;
    int b = idx >> 7;
    const float* ic = input_c + b * 10;
    const float* cw = cnet_w + n * 10;
    float sc = 0.0f;
#pragma unroll
    for (int q = 0; q < 10; ++q) sc += ic[q] * cw[q];
    xc[idx] = sc;
    const float* id = input_d + b * 4;
    float sd = 0.0f;
#pragma unroll
    for (int q = 0; q < 4; ++q) sd += id[q] * theta[q * 128 + n];
    xd[idx] = sd;
}

// ------- prep: W0P pair-interleaved transposed layer-0 weights --------------------
// W0P[((i*K0P2 + k2)*512 + n)*2 + j] = W0_i[n][2*k2 + j]  (zero pad past 17/18)
__global__ void build_w0p(const float* __restrict__ W0_0,
                          const float* __restrict__ W0_rest,
                          float* __restrict__ W0P) {
    int idx = blockIdx.x * 256 + threadIdx.x;      // over 16*K0P2*512*2
    int j  = idx & 1;
    int n  = (idx >> 1) & 511;
    int r  = idx >> 10;
    int k2 = r % K0P2;
    int i  = r / K0P2;
    int k  = 2 * k2 + j;
    float v = 0.0f;
    if (i == 0) {
        if (k < 17) v = W0_0[n * 17 + k];
    } else {
        if (k < 18) v = W0_rest[((i - 1) * 512 + n) * 18 + k];
    }
    W0P[idx] = v;
}

// ------- prep: W1P pair-interleaved transposed layer-1 weights --------------------
// W1P[((i*256 + k2)*512 + n)*2 + j] = W1[i][n][2*k2 + j]
__global__ void build_w1p(const float* __restrict__ W1, float* __restrict__ W1P) {
    size_t idx = (size_t)blockIdx.x * 256 + threadIdx.x; // over 16*512*512
    int j  = (int)(idx & 1);
    int n  = (int)((idx >> 1) & 511);
    size_t r = idx >> 10;
    int k2 = (int)(r & 255);
    int i  = (int)(r >> 8);
    W1P[idx] = W1[((size_t)i * 512 + n) * 512 + 2 * k2 + j];
}

// ---------------- fused step kernel: one of the 16 sequential MLP steps --------------
// block: 64 rows x 512 cols, 256 threads (8 waves of 32)
__global__ __launch_bounds__(256)
void step_kernel(int i,
                 const float* __restrict__ xc, const float* __restrict__ xd,
                 const float* __restrict__ noise, float* __restrict__ prevb,
                 const float* __restrict__ W0P, const float* __restrict__ W1P,
                 const float* __restrict__ b0, const float* __restrict__ b1,
                 const float* __restrict__ W2, const float* __restrict__ b2,
                 float* __restrict__ out) {
    __shared__ float xs[64 * K0PAD];     // staged layer-0 input tile
    __shared__ float h1[64 * H1S];       // relu(layer0) activations, ~132 KB (CDNA5 LDS)
    __shared__ float o_s[64];            // per-row output accumulators

    const int tid   = threadIdx.x;
    const int bbase = blockIdx.x * 64;

    // ---- phase 0: stage x = [ci(8) | di(8) | ni(1) | prev(1) | pad(2)] ----
    for (int idx = tid; idx < 64 * K0PAD; idx += 256) {
        int row = idx / K0PAD;
        int c   = idx - row * K0PAD;
        int b   = bbase + row;
        float v;
        if (c < 8)        v = xc[b * 128 + i * 8 + c];
        else if (c < 16)  v = xd[b * 128 + i * 8 + (c - 8)];
        else if (c == 16) v = noise[b * 16 + i];
        else if (c == 17) v = (i > 0) ? prevb[b] : 0.0f;
        else              v = 0.0f;
        xs[idx] = v;
    }
    if (tid < 64) o_s[tid] = b2[i];
    __syncthreads();

    const int lane     = tid & 31;
    const int wave     = tid >> 5;
    const int l16      = lane & 15;
    const int lhalf    = lane >> 4;      // 0: K={k0,k0+1}/rows 0-7 ; 1: K={k0+2,k0+3}/rows 8-15
    const int koff     = lhalf * 2;
    const int row_tile = wave >> 1;      // 0..3  (16 rows each)
    const int nbase    = (wave & 1) * 256;
    const int arow     = row_tile * 16 + l16;   // A-operand row for this lane
    const int srow     = row_tile * 16 + lhalf * 8;

    // ---- phase 1: h1 = relu(x @ W0^T + b0[i]) via fp32 WMMA ----
    const float* W0Pi = W0P + i * (K0P2 * 512 * 2);
#pragma unroll
    for (int t = 0; t < 16; ++t) {
        const int n = nbase + t * 16 + l16;
        const float bias = b0[i * 512 + n];
        v8f acc;
#pragma unroll
        for (int r = 0; r < 8; ++r) acc[r] = bias;
#pragma unroll
        for (int k0 = 0; k0 < K0PAD; k0 += 4) {
            v2f a = *(const v2f*)&xs[arow * K0PAD + k0 + koff];
            // pair-interleaved: {W0T[k][n], W0T[k+1][n]} contiguous, k = k0 + koff
            v2f bb = *(const v2f*)&W0Pi[((k0 / 2 + lhalf) * 512 + n) * 2];
            acc = wmma_f32_4(a, bb, acc);
        }
#pragma unroll
        for (int r = 0; r < 8; ++r)
            h1[(srow + r) * H1S + n] = fmaxf(acc[r], 0.0f);
    }
    __syncthreads();

    // ---- phase 2: h2 = relu(h1 @ W1^T + b1[i]) fused with o = h2 . w2 ----
    v8f acc[16];
#pragma unroll
    for (int t = 0; t < 16; ++t) {
        const float bias = b1[i * 512 + nbase + t * 16 + l16];
#pragma unroll
        for (int r = 0; r < 8; ++r) acc[t][r] = bias;
    }
    const float* W1Pi = W1P + (size_t)i * 512 * 512;
    for (int k0 = 0; k0 < 512; k0 += 4) {
        v2f a = *(const v2f*)&h1[arow * H1S + k0 + koff];
        // base of this lane's B pairs for k = k0 + koff
        const float* Wp = W1Pi + ((size_t)(k0 / 2 + lhalf) * 512 + nbase + l16) * 2;
#pragma unroll
        for (int t = 0; t < 16; ++t) {
            v2f bb = *(const v2f*)&Wp[t * 32];   // one coalesced b64 per tile
            acc[t] = wmma_f32_4(a, bb, acc[t]);
        }
    }
    // fold layer 2: relu then dot with w2 (per-lane partials over this wave's 256 cols)
    float rs[8];
#pragma unroll
    for (int r = 0; r < 8; ++r) rs[r] = 0.0f;
#pragma unroll
    for (int t = 0; t < 16; ++t) {
        const float w2v = W2[i * 512 + nbase + t * 16 + l16];
#pragma unroll
        for (int r = 0; r < 8; ++r)
            rs[r] += fmaxf(acc[t][r], 0.0f) * w2v;
    }
    // reduce across the 16 lanes holding different columns of the same rows
#pragma unroll
    for (int m = 1; m < 16; m <<= 1) {
#pragma unroll
        for (int r = 0; r < 8; ++r)
            rs[r] += __shfl_xor(rs[r], m, 32);
    }
    if (l16 == 0) {
#pragma unroll
        for (int r = 0; r < 8; ++r)
            atomicAdd(&o_s[row_tile * 16 + lhalf * 8 + r], rs[r]);
    }
    __syncthreads();

    // ---- phase 3: write output column i and prev ----
    if (tid < 64) {
        float o = o_s[tid];
        out[(size_t)(bbase + tid) * 16 + i] = o;
        prevb[bbase + tid] = o;
    }
}

// ---------------- host launcher ----------------
extern "C" void kernel_launch(void* const* d_in, const int* in_sizes, int n_in,
                              void* d_out, int out_size, void* d_ws, size_t ws_size,
                              hipStream_t stream) {
    const float* noise   = (const float*)d_in[0];
    const float* input_c = (const float*)d_in[1];
    const float* input_d = (const float*)d_in[2];
    const float* noise_d = (const float*)d_in[3];
    const float* cnet_w  = (const float*)d_in[4];
    const float* mu      = (const float*)d_in[5];
    const float* sigma   = (const float*)d_in[6];
    const float* W0_0    = (const float*)d_in[7];
    const float* W0_rest = (const float*)d_in[8];
    const float* b0      = (const float*)d_in[9];
    const float* W1      = (const float*)d_in[10];
    const float* b1      = (const float*)d_in[11];
    const float* W2      = (const float*)d_in[12];
    const float* b2      = (const float*)d_in[13];

    float* out = (float*)d_out;
    float* kl_out = out + (size_t)BQ * IDIM;

    // workspace carve-up (floats)
    float* ws    = (float*)d_ws;
    float* theta = ws;                                   // 4*128
    float* xc    = theta + 512;                          // B*128
    float* xd    = xc + (size_t)BQ * 128;                // B*128
    float* prevb = xd + (size_t)BQ * 128;                // B
    float* W0P   = prevb + BQ;                           // 16*K0P2*512*2
    float* W1P   = W0P + (size_t)IDIM * K0P2 * 512 * 2;  // 16*512*512

    prep_theta_kl<<<1, 512, 0, stream>>>(mu, sigma, noise_d, theta, kl_out);
    prep_inputs<<<(BQ * 128) / 256, 256, 0, stream>>>(input_c, input_d, cnet_w, theta, xc, xd);
    build_w0p<<<(IDIM * K0P2 * 512 * 2) / 256, 256, 0, stream>>>(W0_0, W0_rest, W0P);
    build_w1p<<<(IDIM * 512 * 512) / 256, 256, 0, stream>>>(W1, W1P);

    for (int i = 0; i < IDIM; ++i) {
        step_kernel<<<BQ / 64, 256, 0, stream>>>(i, xc, xd, noise, prevb,
                                                 W0P, W1P, b0, b1, W2, b2, out);
    }
}